// CustomTransformerEncoder_58402965291267
// MI455X (gfx1250) — compile-verified
//
#include <hip/hip_runtime.h>
#include <math.h>

// ---- problem constants (match reference) ----
#define LNUM 6
#define BB   4
#define SSQ  1024
#define DDM  256
#define HHD  8
#define DHH  32          // head dim
#define FFF  128
#define NTOK (BB*SSQ)    // 4096 tokens
#define QKVW (3*DDM)     // 768
#define ATTN_SCALE 0.17677669529663687f  // 1/sqrt(32)

typedef __attribute__((ext_vector_type(2))) float v2f;
typedef __attribute__((ext_vector_type(4))) float v4f;
typedef __attribute__((ext_vector_type(8))) float v8f;

// vector-of-4-int pointer types matching the async-LDS builtin signature
typedef int vi4 __attribute__((vector_size(4 * sizeof(int))));
typedef __attribute__((address_space(1))) vi4 gas_vi4;   // global
typedef __attribute__((address_space(3))) vi4 las_vi4;   // LDS

// ---- gfx1250 async global->LDS path (guarded; falls back to ds stores) ----
#if defined(__has_builtin)
#  if __has_builtin(__builtin_amdgcn_global_load_async_to_lds_b128) && \
      __has_builtin(__builtin_amdgcn_s_wait_asynccnt)
#    define USE_ASYNC_LDS 1
#  endif
#endif
#ifndef USE_ASYNC_LDS
#  define USE_ASYNC_LDS 0
#endif

__device__ __forceinline__ v8f wmma4(v2f a, v2f b, v8f c) {
  // V_WMMA_F32_16X16X4_F32 : D = A(16x4) * B(4x16) + C(16x16)
  return __builtin_amdgcn_wmma_f32_16x16x4_f32(
      /*neg_a=*/false, a, /*neg_b=*/false, b,
      /*c_mod=*/(short)0, c, /*reuse_a=*/false, /*reuse_b=*/false);
}

__device__ __forceinline__ v8f zero8() {
  v8f z = {0.f,0.f,0.f,0.f,0.f,0.f,0.f,0.f};
  return z;
}

// ---------------------------------------------------------------------------
// Generic row-major GEMM: C[M,N] = alpha * A[M,K] @ B[K,N] (+bias[n]) (+resid)
// block = 128 threads (4 waves), each wave -> 16 rows x 64 cols of C.
// The 64-wide B panel is identical for all 4 waves -> stage it in LDS once per
// 64-row K-chunk (async global->LDS copy when available).
// LDS layout: Bs[k][n], row stride 80 floats (banks of rows k and k+2 disjoint;
// 320B rows keep 16B alignment for b128 async writes).
// grid = (N/64, M/64). All dims multiples of tile sizes.
// ---------------------------------------------------------------------------
#define BS_STRIDE 80
#define KCHUNK    64

template<int TN>
__global__ void __launch_bounds__(128)
gemm_rm(const float* __restrict__ A, const float* __restrict__ B,
        const float* __restrict__ bias, const float* __restrict__ resid,
        float* __restrict__ C, int K, int lda, int ldb, int ldc, float alpha)
{
  __shared__ float Bs[KCHUNK * BS_STRIDE];          // 20 KB

  const int lane  = threadIdx.x & 31;
  const int wave  = threadIdx.x >> 5;
  const int mBase = blockIdx.y * 64 + wave * 16;
  const int nBase = blockIdx.x * (TN * 16);
  const int l15   = lane & 15;
  const int koff  = (lane >> 4) << 1;   // 0 or 2 (f32 A/B fragment layout)

  // cooperative-load lane mapping: 2 rows per wave-instruction
  const int ldRow = 2 * wave + (lane >> 4);         // 0..7
  const int ldCol = l15 * 4;                        // 16B per lane

  v8f acc[TN];
  #pragma unroll
  for (int t = 0; t < TN; ++t) acc[t] = zero8();

  const float* arow = A + (long)(mBase + l15) * lda + koff;

  for (int kc = 0; kc < K; kc += KCHUNK) {
    __syncthreads();   // previous chunk fully consumed before overwrite

    // ---- stage B[kc .. kc+63][nBase .. nBase+63] into LDS ----
    #pragma unroll
    for (int r0 = 0; r0 < KCHUNK; r0 += 8) {
      const int r = r0 + ldRow;
      const float* gp = B + (long)(kc + r) * ldb + nBase + ldCol;
      float* lp = &Bs[r * BS_STRIDE + ldCol];
#if USE_ASYNC_LDS
      __builtin_amdgcn_global_load_async_to_lds_b128(
          (gas_vi4*)gp, (las_vi4*)lp, 0, 0);
#else
      *(v4f*)lp = *(const v4f*)gp;
#endif
    }
#if USE_ASYNC_LDS
    __builtin_amdgcn_s_wait_asynccnt(0);
#endif
    __syncthreads();   // all waves' panel writes visible

    // ---- consume chunk ----
    #pragma unroll
    for (int k0 = 0; k0 < KCHUNK; k0 += 4) {
      if ((k0 & 31) == 0)  // uniform prefetch of upcoming A -> global_prefetch_b8
        __builtin_prefetch(arow + kc + k0 + 64, 0, 1);
      v2f a = *(const v2f*)(arow + kc + k0);
      const float* bp = &Bs[(k0 + koff) * BS_STRIDE + l15];
      #pragma unroll
      for (int t = 0; t < TN; ++t) {
        v2f b;
        b.x = bp[t * 16];
        b.y = bp[t * 16 + BS_STRIDE];
        acc[t] = wmma4(a, b, acc[t]);
      }
    }
  }

  const int hiM = (lane >> 4) << 3;     // C layout: row = r + 8*(lane>=16)
  #pragma unroll
  for (int r = 0; r < 8; ++r) {
    const int m = mBase + hiM + r;
    #pragma unroll
    for (int t = 0; t < TN; ++t) {
      const int n = nBase + t * 16 + l15;
      float v = acc[t][r] * alpha;
      if (bias)  v += bias[n];
      const long ci = (long)m * ldc + n;
      if (resid) v += resid[ci];
      C[ci] = v;
    }
  }
}

// ---------------------------------------------------------------------------
// Attention logits: for bh = blockIdx.z, aw[bh, m, n] = scale * Q[m,:] . K[n,:]
// Q rows at qkv[tok*768 + h*32 + k], K rows at qkv[tok*768 + 256 + h*32 + k]
// ---------------------------------------------------------------------------
__global__ void __launch_bounds__(128)
attn_logits(const float* __restrict__ qkv, float* __restrict__ aw)
{
  const int lane  = threadIdx.x & 31;
  const int wave  = threadIdx.x >> 5;
  const int bh    = blockIdx.z;
  const int b     = bh >> 3, h = bh & 7;
  const int mBase = blockIdx.y * 64 + wave * 16;
  const int nBase = blockIdx.x * 64;
  const int l15   = lane & 15;
  const int koff  = (lane >> 4) << 1;

  const float* Q  = qkv + (long)(b * SSQ) * QKVW + h * DHH;
  const float* Kb = qkv + (long)(b * SSQ) * QKVW + DDM + h * DHH;

  v8f acc[4];
  #pragma unroll
  for (int t = 0; t < 4; ++t) acc[t] = zero8();

  const float* qrow = Q + (long)(mBase + l15) * QKVW + koff;
  #pragma unroll
  for (int k0 = 0; k0 < DHH; k0 += 4) {
    v2f a = *(const v2f*)(qrow + k0);
    #pragma unroll
    for (int t = 0; t < 4; ++t) {
      const int n = nBase + t * 16 + l15;
      v2f bf = *(const v2f*)(Kb + (long)n * QKVW + k0 + koff); // K^T: contiguous in k
      acc[t] = wmma4(a, bf, acc[t]);
    }
  }

  float* out = aw + (long)bh * SSQ * SSQ;
  const int hiM = (lane >> 4) << 3;
  #pragma unroll
  for (int r = 0; r < 8; ++r) {
    const long m = mBase + hiM + r;
    #pragma unroll
    for (int t = 0; t < 4; ++t)
      out[m * SSQ + nBase + t * 16 + l15] = acc[t][r] * ATTN_SCALE;
  }
}

// ---------------------------------------------------------------------------
// Row softmax over 1024 elements, in place. block = 256, 4 elems/thread.
// ---------------------------------------------------------------------------
__global__ void __launch_bounds__(256)
softmax_rows(float* __restrict__ aw)
{
  __shared__ float red[256];
  float* p = aw + (long)blockIdx.x * SSQ;
  const int tid = threadIdx.x;

  float v0 = p[tid], v1 = p[tid + 256], v2 = p[tid + 512], v3 = p[tid + 768];
  red[tid] = fmaxf(fmaxf(v0, v1), fmaxf(v2, v3));
  __syncthreads();
  for (int s = 128; s > 0; s >>= 1) {
    if (tid < s) red[tid] = fmaxf(red[tid], red[tid + s]);
    __syncthreads();
  }
  const float mx = red[0];
  __syncthreads();

  v0 = expf(v0 - mx); v1 = expf(v1 - mx); v2 = expf(v2 - mx); v3 = expf(v3 - mx);
  red[tid] = v0 + v1 + v2 + v3;
  __syncthreads();
  for (int s = 128; s > 0; s >>= 1) {
    if (tid < s) red[tid] += red[tid + s];
    __syncthreads();
  }
  const float inv = 1.0f / red[0];
  p[tid]       = v0 * inv;
  p[tid + 256] = v1 * inv;
  p[tid + 512] = v2 * inv;
  p[tid + 768] = v3 * inv;
}

// ---------------------------------------------------------------------------
// o = aw @ V : per bh, M=1024, N=32, K=1024.
// V[k][n] at qkv[(b*1024+k)*768 + 512 + h*32 + n]; out into obuf[tok*256+h*32+n]
// ---------------------------------------------------------------------------
__global__ void __launch_bounds__(128)
attn_av(const float* __restrict__ aw, const float* __restrict__ qkv,
        float* __restrict__ obuf)
{
  const int lane  = threadIdx.x & 31;
  const int wave  = threadIdx.x >> 5;
  const int bh    = blockIdx.z;
  const int b     = bh >> 3, h = bh & 7;
  const int mBase = blockIdx.y * 64 + wave * 16;
  const int l15   = lane & 15;
  const int koff  = (lane >> 4) << 1;

  const float* Arow = aw + (long)bh * SSQ * SSQ + (long)(mBase + l15) * SSQ + koff;
  const float* V    = qkv + (long)(b * SSQ) * QKVW + 2 * DDM + h * DHH;

  v8f acc[2];
  acc[0] = zero8(); acc[1] = zero8();

  for (int k0 = 0; k0 < SSQ; k0 += 4) {
    v2f a = *(const v2f*)(Arow + k0);
    #pragma unroll
    for (int t = 0; t < 2; ++t) {
      const float* bp = V + (long)(k0 + koff) * QKVW + t * 16 + l15;
      v2f bf;
      bf.x = bp[0];
      bf.y = bp[QKVW];
      acc[t] = wmma4(a, bf, acc[t]);
    }
  }

  const int hiM = (lane >> 4) << 3;
  #pragma unroll
  for (int r = 0; r < 8; ++r) {
    const long tok = (long)b * SSQ + mBase + hiM + r;
    #pragma unroll
    for (int t = 0; t < 2; ++t)
      obuf[tok * DDM + h * DHH + t * 16 + l15] = acc[t][r];
  }
}

// ---------------------------------------------------------------------------
// LayerNorm over D=256; one block per token.
// ---------------------------------------------------------------------------
__global__ void __launch_bounds__(256)
layernorm(const float* __restrict__ x, const float* __restrict__ w,
          const float* __restrict__ bb, float* __restrict__ out)
{
  __shared__ float red[256];
  const int tid = threadIdx.x;
  const long base = (long)blockIdx.x * DDM;
  const float v = x[base + tid];

  red[tid] = v;
  __syncthreads();
  for (int s = 128; s > 0; s >>= 1) {
    if (tid < s) red[tid] += red[tid + s];
    __syncthreads();
  }
  const float mean = red[0] * (1.0f / DDM);
  __syncthreads();

  const float d = v - mean;
  red[tid] = d * d;
  __syncthreads();
  for (int s = 128; s > 0; s >>= 1) {
    if (tid < s) red[tid] += red[tid + s];
    __syncthreads();
  }
  const float var = red[0] * (1.0f / DDM);
  out[base + tid] = d * rsqrtf(var + 1e-5f) * w[tid] + bb[tid];
}

// ---------------------------------------------------------------------------
// GEGLU: act[tok,j] = h[tok,j] * gelu_exact(h[tok,128+j]),  j in [0,128)
// ---------------------------------------------------------------------------
__global__ void __launch_bounds__(256)
geglu(const float* __restrict__ h, float* __restrict__ act)
{
  const long i   = (long)blockIdx.x * 256 + threadIdx.x;  // NTOK*FFF elements
  const long row = i >> 7;
  const long col = i & 127;
  const float a = h[row * 256 + col];
  const float g = h[row * 256 + 128 + col];
  act[i] = a * (0.5f * g * (1.0f + erff(g * 0.70710678118654752f)));
}

// ---------------------------------------------------------------------------
// host side
// ---------------------------------------------------------------------------
extern "C" void kernel_launch(void* const* d_in, const int* in_sizes, int n_in,
                              void* d_out, int out_size, void* d_ws, size_t ws_size,
                              hipStream_t stream)
{
  const float* x_in = (const float*)d_in[0];
  const float* ln1w = (const float*)d_in[1];
  const float* ln1b = (const float*)d_in[2];
  const float* wqkv = (const float*)d_in[3];
  const float* wout = (const float*)d_in[4];
  const float* ln2w = (const float*)d_in[5];
  const float* ln2b = (const float*)d_in[6];
  const float* w1   = (const float*)d_in[7];
  const float* b1   = (const float*)d_in[8];
  const float* w2   = (const float*)d_in[9];
  const float* b2   = (const float*)d_in[10];

  float* out    = (float*)d_out;
  float* xcur   = out;                           // [B,S,D] final x lives here
  float* aw_all = out + (long)NTOK * DDM;        // [L,B,H,S,S]

  float* ws   = (float*)d_ws;
  float* xn   = ws;                              // [4096,256]
  float* qkv  = xn   + (long)NTOK * DDM;         // [4096,768]
  float* obuf = qkv  + (long)NTOK * QKVW;        // [4096,256]
  float* hbuf = obuf + (long)NTOK * DDM;         // [4096,256]
  float* act  = hbuf + (long)NTOK * DDM;         // [4096,128]

  // x working copy directly in d_out
  (void)hipMemcpyAsync(xcur, x_in, (size_t)NTOK * DDM * sizeof(float),
                       hipMemcpyDeviceToDevice, stream);

  const dim3 blk128(128), blk256(256);

  for (int i = 0; i < LNUM; ++i) {
    // ---- attention block ----
    layernorm<<<NTOK, blk256, 0, stream>>>(xcur, ln1w + i * DDM, ln1b + i * DDM, xn);

    // qkv = xn @ wqkv_i   [4096,256]x[256,768]
    gemm_rm<4><<<dim3(QKVW / 64, NTOK / 64), blk128, 0, stream>>>(
        xn, wqkv + (long)i * DDM * QKVW, nullptr, nullptr, qkv,
        DDM, DDM, QKVW, QKVW, 1.0f);

    float* awL = aw_all + (long)i * BB * HHD * SSQ * SSQ;
    attn_logits<<<dim3(16, 16, BB * HHD), blk128, 0, stream>>>(qkv, awL);
    softmax_rows<<<BB * HHD * SSQ, blk256, 0, stream>>>(awL);
    attn_av<<<dim3(1, 16, BB * HHD), blk128, 0, stream>>>(awL, qkv, obuf);

    // x += o @ wout_i  (in-place residual)
    gemm_rm<4><<<dim3(DDM / 64, NTOK / 64), blk128, 0, stream>>>(
        obuf, wout + (long)i * DDM * DDM, nullptr, xcur, xcur,
        DDM, DDM, DDM, DDM, 1.0f);

    // ---- GEGLU FFN ----
    layernorm<<<NTOK, blk256, 0, stream>>>(xcur, ln2w + i * DDM, ln2b + i * DDM, xn);

    // h = xn @ w1_i + b1_i   [4096,256]x[256,256]
    gemm_rm<4><<<dim3(256 / 64, NTOK / 64), blk128, 0, stream>>>(
        xn, w1 + (long)i * DDM * 256, b1 + (long)i * 256, nullptr, hbuf,
        DDM, DDM, 256, 256, 1.0f);

    geglu<<<(NTOK * FFF) / 256, blk256, 0, stream>>>(hbuf, act);

    // x += act @ w2_i + b2_i   [4096,128]x[128,256]
    gemm_rm<4><<<dim3(DDM / 64, NTOK / 64), blk128, 0, stream>>>(
        act, w2 + (long)i * FFF * DDM, b2 + (long)i * DDM, xcur, xcur,
        FFF, FFF, DDM, DDM, 1.0f);
  }
}